// SpatioTemporalRegressor_10333691314545
// MI455X (gfx1250) — compile-verified
//
#include <hip/hip_runtime.h>
#include <math.h>

// ---------------------------------------------------------------------------
// SpatioTemporalRegressor on MI455X (gfx1250, wave32).
// Memory-bound: streams 164MB dgraph once (non-temporal); WMMA f32 16x16x4
// (A=ones) performs the j-reduction on the CDNA5 matrix pipe. All hot-loop
// addressing is scalar-base + 32-bit voffset (no per-iter 64-bit math).
// ---------------------------------------------------------------------------

typedef __attribute__((ext_vector_type(2))) float v2f;
typedef __attribute__((ext_vector_type(8))) float v8f;

#if defined(__has_builtin)
#if __has_builtin(__builtin_amdgcn_wmma_f32_16x16x4_f32)
#define USE_WMMA_F32X4 1
#endif
#endif

#define MEMD 20

__global__ void init_ws_kernel(float* __restrict__ loss_acc,
                               float* __restrict__ alpha, int n) {
    int idx = blockIdx.x * blockDim.x + threadIdx.x;
    if (idx == 0) *loss_acc = 0.0f;
    if (idx < n) alpha[idx] = 0.0f;
}

__global__ void scatter_alpha_kernel(const int* __restrict__ coords,
                                     const float* __restrict__ anz,
                                     float* __restrict__ alpha, int K, int NNZ) {
    int i = blockIdx.x * blockDim.x + threadIdx.x;
    if (i < NNZ) {
        int r = coords[i];
        int c = coords[NNZ + i];
        atomicAdd(&alpha[r * K + c], anz[i]);
    }
}

// one block per k: base[k] = mean_t(speeds[t,k]) / 5 ; broadcast into pred0 row
__global__ void base_pred0_kernel(const float* __restrict__ speeds,
                                  float* __restrict__ base,
                                  float* __restrict__ out_pred0, int K, int T) {
    __shared__ float sdata[256];
    __shared__ float sbase;
    const int k = blockIdx.x;
    float acc = 0.0f;
    for (int t = threadIdx.x; t < T; t += blockDim.x)
        acc += speeds[t * K + k];
    sdata[threadIdx.x] = acc;
    __syncthreads();
    for (int s = blockDim.x / 2; s > 0; s >>= 1) {
        if (threadIdx.x < (unsigned)s) sdata[threadIdx.x] += sdata[threadIdx.x + s];
        __syncthreads();
    }
    if (threadIdx.x == 0) {
        float b = (sdata[0] / (float)T) * 0.2f;
        base[k] = b;
        sbase = b;
    }
    __syncthreads();
    float b = sbase;
    for (int t = threadIdx.x; t < T; t += blockDim.x)
        out_pred0[(size_t)k * T + t] = b;
}

// h[k,t] = sum_{lag=1..D} Beta[k]*exp(-lag*Beta[k]) * S[k,t-lag]; stored [T,K]
__global__ void h_kernel(const float* __restrict__ speeds,
                         const float* __restrict__ Beta,
                         float* __restrict__ ht, int K, int T) {
    int idx = blockIdx.x * blockDim.x + threadIdx.x;
    if (idx >= K * T) return;
    int k = idx % K;
    int t = idx / K;
    float beta  = Beta[k];
    float decay = expf(-beta);
    float w     = beta * decay;  // lag = 1 weight
    float acc   = 0.0f;
    for (int lag = 1; lag <= MEMD; ++lag) {
        int tp = t - lag;
        if (tp >= 0) acc += w * speeds[tp * K + k];
        w *= decay;
    }
    ht[(size_t)t * K + k] = acc;
}

__device__ __forceinline__ float softplusf(float x) {
    if (x > 20.0f)  return x;
    if (x < -20.0f) return expf(x);
    return log1pf(expf(x));
}

// 1 wave per (t, 16-wide k tile). j-reduction via V_WMMA_F32_16X16X4_F32 with
// A = ones: D[m,n] += sum over the 4 B rows of column n.
// Lanes 0-15 supply rows (j, j+1), lanes 16-31 rows (j+2, j+3); column = lane%16.
__global__ void __launch_bounds__(32)
pred1_loss_kernel(const float* __restrict__ dgraph,
                  const float* __restrict__ alpha,
                  const float* __restrict__ ht,
                  const float* __restrict__ speeds,
                  const float* __restrict__ base,
                  float* __restrict__ out_pred1,
                  float* __restrict__ loss_acc, int K, int T) {
    const int t     = blockIdx.y;
    const int kbase = blockIdx.x * 16;
    const int lane  = threadIdx.x;
    const int col   = lane & 15;
    const int half  = lane >> 4;
    const int k     = kbase + col;
    const int kc    = (k < K) ? k : (K - 1);
    const float km  = (k < K) ? 1.0f : 0.0f;

    float s = 0.0f;
    if (t != 0) {                                  // uniform branch: EXEC all-1
        const float* __restrict__ dgt = dgraph + (size_t)t * K * K;  // scalar base
        const float* __restrict__ htt = ht + (size_t)t * K;
        const int njg  = K >> 2;                   // full groups of 4 j's
        const int step = 4 * K;

        // 32-bit offsets into alpha/dgt (both [K,K] slices) and htt
        int offA = (half * 2) * K + kc;
        int offH = half * 2;

#ifdef USE_WMMA_F32X4
        v8f acc = {};
        v2f a;  a[0] = 1.0f; a[1] = 1.0f;          // ones A-matrix
        for (int jg = 0; jg < njg; ++jg) {
            float d0 = __builtin_nontemporal_load(dgt + offA);
            float d1 = __builtin_nontemporal_load(dgt + offA + K);
            float e0 = km * alpha[offA]     * htt[offH]     * d0;
            float e1 = km * alpha[offA + K] * htt[offH + 1] * d1;
            v2f b;  b[0] = e0; b[1] = e1;
            acc = __builtin_amdgcn_wmma_f32_16x16x4_f32(
                false, a, false, b, (short)0, acc, false, false);
            offA += step;
            offH += 4;
        }
        if (K & 3) {                               // masked tail group
            int jA  = njg * 4 + half * 2;
            int jB  = jA + 1;
            int jAc = (jA < K) ? jA : (K - 1);
            int jBc = (jB < K) ? jB : (K - 1);
            float mA = (jA < K) ? km : 0.0f;
            float mB = (jB < K) ? km : 0.0f;
            float e0 = mA * alpha[jAc * K + kc] * htt[jAc]
                          * __builtin_nontemporal_load(dgt + jAc * K + kc);
            float e1 = mB * alpha[jBc * K + kc] * htt[jBc]
                          * __builtin_nontemporal_load(dgt + jBc * K + kc);
            v2f b;  b[0] = e0; b[1] = e1;
            acc = __builtin_amdgcn_wmma_f32_16x16x4_f32(
                false, a, false, b, (short)0, acc, false, false);
        }
        s = acc[0];                                // row M=0 (all rows equal)
#else
        float sacc = 0.0f;
        for (int jg = 0; jg < njg; ++jg) {
            float d0 = __builtin_nontemporal_load(dgt + offA);
            float d1 = __builtin_nontemporal_load(dgt + offA + K);
            sacc += km * alpha[offA]     * htt[offH]     * d0;
            sacc += km * alpha[offA + K] * htt[offH + 1] * d1;
            offA += step;
            offH += 4;
        }
        if (K & 3) {
            int jA  = njg * 4 + half * 2;
            int jB  = jA + 1;
            int jAc = (jA < K) ? jA : (K - 1);
            int jBc = (jB < K) ? jB : (K - 1);
            float mA = (jA < K) ? km : 0.0f;
            float mB = (jB < K) ? km : 0.0f;
            sacc += mA * alpha[jAc * K + kc] * htt[jAc]
                       * __builtin_nontemporal_load(dgt + jAc * K + kc);
            sacc += mB * alpha[jBc * K + kc] * htt[jBc]
                       * __builtin_nontemporal_load(dgt + jBc * K + kc);
        }
        s = sacc + __shfl_xor(sacc, 16);
#endif
    }

    float pred1 = (t == 0) ? 0.0f : softplusf(s);

    float c = 0.0f;
    if (k < K && half == 0) {
        out_pred1[(size_t)k * T + t] = pred1;
        float diff = speeds[t * K + k] - pred1 - base[k];
        c = diff * diff;
    }
    // full-wave tree reduction (wave32)
    for (int off = 16; off > 0; off >>= 1) c += __shfl_down(c, off);
    if (lane == 0) atomicAdd(loss_acc, c);
}

__global__ void finalize_kernel(const float* __restrict__ loss_acc,
                                float* __restrict__ out, int K, int T) {
    if (blockIdx.x == 0 && threadIdx.x == 0)
        out[0] = loss_acc[0] / (float)(K * T);
}

extern "C" void kernel_launch(void* const* d_in, const int* in_sizes, int n_in,
                              void* d_out, int out_size, void* d_ws, size_t ws_size,
                              hipStream_t stream) {
    const float* speeds = (const float*)d_in[0];   // [T,K]
    const float* dgraph = (const float*)d_in[1];   // [T,K,K]
    const int*   coords = (const int*)d_in[2];     // [2,NNZ]
    const float* anz    = (const float*)d_in[3];   // [NNZ]
    const float* Beta   = (const float*)d_in[4];   // [K]

    const int K   = in_sizes[4];
    const int T   = in_sizes[0] / K;
    const int NNZ = in_sizes[3];

    float* out      = (float*)d_out;
    float* ws       = (float*)d_ws;
    float* loss_acc = ws;                          // 1 float
    float* base     = ws + 16;                     // K floats
    float* alpha    = ws + 256;                    // K*K floats
    float* ht       = alpha + (size_t)K * K;       // T*K floats
    float* pred0    = out + 1;                     // [K,T]
    float* pred1    = out + 1 + (size_t)K * T;     // [K,T]

    const int nA = K * K;
    init_ws_kernel<<<(nA + 255) / 256, 256, 0, stream>>>(loss_acc, alpha, nA);
    scatter_alpha_kernel<<<(NNZ + 255) / 256, 256, 0, stream>>>(coords, anz, alpha, K, NNZ);
    base_pred0_kernel<<<K, 256, 0, stream>>>(speeds, base, pred0, K, T);
    h_kernel<<<(K * T + 255) / 256, 256, 0, stream>>>(speeds, Beta, ht, K, T);

    dim3 grid((K + 15) / 16, T);
    pred1_loss_kernel<<<grid, 32, 0, stream>>>(dgraph, alpha, ht, speeds, base,
                                               pred1, loss_acc, K, T);
    finalize_kernel<<<1, 1, 0, stream>>>(loss_acc, out, K, T);
}